// PatternConv_70128226009239
// MI455X (gfx1250) — compile-verified
//
#include <hip/hip_runtime.h>
#include <hip/hip_bf16.h>

typedef float v2f __attribute__((ext_vector_type(2)));
typedef float v4f __attribute__((ext_vector_type(4)));
typedef float v8f __attribute__((ext_vector_type(8)));

#define NCOUT 192   // output channels
#define NH    256
#define NW    256

// One workgroup: 256 threads = 8 wave32, wave index == w (kernel col phase).
// Block fixes (h, ntile) -> weights held in registers; iterates 8 M-tiles
// (b, p, qhalf), each producing 16 channels x 128 contiguous output columns.
__global__ void __launch_bounds__(256) pattern_conv_wmma(
    const float* __restrict__ x,      // (4, 1, 256, 256)
    const float* __restrict__ wgt,    // (8, 8, 192, 1, 8, 8)
    const float* __restrict__ bias,   // (8, 8, 192)
    float* __restrict__ out)          // (4, 192, 256, 256)
{
    int bid    = blockIdx.x;
    int ntile  = bid % 12;  bid /= 12;
    int h      = bid % 8;   bid /= 8;
    int mchunk = bid;                      // 0..31, selects 8 of 256 M-tiles

    const int tid  = threadIdx.x;
    const int w    = tid >> 5;             // wave id == kernel col phase 0..7
    const int lane = tid & 31;
    const int lh   = lane >> 4;            // lane half (K MSB for A/B, M MSB for C/D)
    const int ln   = lane & 15;
    const int n0   = ntile * 16;           // channel tile base

    __shared__ float in_lds[8][144];       // 8 x 135 used (zero-padded OOB)
    __shared__ float out_lds[16][128];     // 16 channels x 128 contiguous cols

    // ---- A matrix (weights), resident for all 8 M-tiles ----
    // A VGPR layout (16x4 f32): lane ln holds M=ln; chunk t, vgpr v -> K = 4t + lh*2 + v
    const float* wbase =
        wgt + (((size_t)h * 8 + w) * NCOUT + (n0 + ln)) * 64 + lh * 2;
    v2f A[16];
#pragma unroll
    for (int t = 0; t < 16; ++t) {
        A[t].x = wbase[4 * t + 0];
        A[t].y = wbase[4 * t + 1];
    }

    // ---- bias -> accumulator seed: vgpr g, lane -> M = g + lh*8 (channel) ----
    v8f Cinit;
    {
        const v4f* bv = (const v4f*)(bias + ((size_t)h * 8 + w) * NCOUT + n0 + lh * 8);
        v4f b0 = bv[0], b1 = bv[1];
#pragma unroll
        for (int g = 0; g < 4; ++g) { Cinit[g] = b0[g]; Cinit[g + 4] = b1[g]; }
    }

    for (int it = 0; it < 8; ++it) {
        const int m     = mchunk * 8 + it;     // 0..255
        const int qhalf = m & 1;
        const int p     = (m >> 1) & 31;
        const int b     = m >> 6;

        const int row0 = p * 8 + h;            // output row; input rows row0..row0+7
        const int col0 = qhalf * 128;          // col base; need cols col0..col0+134

        // ---- stage input tile into LDS (v4f chunks; pad boundary is 16B-aligned) ----
        for (int idx = tid; idx < 8 * 36; idx += 256) {
            int i  = idx / 36;
            int c4 = (idx - i * 36) * 4;
            int r  = row0 + i;
            int cc = col0 + c4;
            v4f v  = {0.0f, 0.0f, 0.0f, 0.0f};
            if (r < NH && cc < NW)
                v = *(const v4f*)(x + ((size_t)b * NH + r) * NW + cc);
            *(v4f*)&in_lds[i][c4] = v;

            // speculative prefetch of next M-tile's input (64B granularity)
            if (it + 1 < 8 && (idx & 3) == 0) {
                int mn = m + 1;
                int rn = ((mn >> 1) & 31) * 8 + h + i;
                int cn = (mn & 1) * 128 + c4;
                if (rn < NH && cn < NW)
                    __builtin_prefetch(x + ((size_t)(mn >> 6) * NH + rn) * NW + cn, 0, 1);
            }
        }
        __syncthreads();

        // ---- K loop: 16 chained V_WMMA_F32_16X16X4_F32 over k = i*8 + j ----
        // B VGPR layout (4x16 f32): lane ln holds N=ln; chunk t, vgpr v -> K = 4t + lh*2 + v
        // B[k][n] = x[b, row0 + i, col0 + n*8 + w + j],  k = i*8 + j
        v8f C = Cinit;
#pragma unroll
        for (int t = 0; t < 16; ++t) {
            int i = t >> 1;
            int j = (t & 1) * 4 + lh * 2;
            int c = ln * 8 + w + j;
            v2f Bv;
            Bv.x = in_lds[i][c];
            Bv.y = in_lds[i][c + 1];
            C = __builtin_amdgcn_wmma_f32_16x16x4_f32(
                    /*neg_a=*/false, A[t], /*neg_b=*/false, Bv,
                    /*c_mod=*/(short)0, C, /*reuse_a=*/false, /*reuse_b=*/false);
        }

        // ---- scatter D: row = channel m, col = ln*8 + w (disjoint per wave) ----
#pragma unroll
        for (int g = 0; g < 8; ++g)
            out_lds[g + lh * 8][ln * 8 + w] = C[g];

        __syncthreads();

        // ---- coalesced b128 stores: 16 channels x 128 contiguous columns ----
        {
            int r   = tid >> 4;                 // 0..15 (channel row)
            int cl0 = (tid & 15) * 8;           // 0..120
            v4f o0 = *(const v4f*)&out_lds[r][cl0];
            v4f o1 = *(const v4f*)&out_lds[r][cl0 + 4];
            float* op = out + (((size_t)b * NCOUT + n0 + r) * NH + row0) * NW + col0 + cl0;
            *(v4f*)op       = o0;
            *(v4f*)(op + 4) = o1;
        }
        __syncthreads();   // protect in_lds/out_lds reuse next iteration
    }
}

extern "C" void kernel_launch(void* const* d_in, const int* in_sizes, int n_in,
                              void* d_out, int out_size, void* d_ws, size_t ws_size,
                              hipStream_t stream) {
    const float* x    = (const float*)d_in[0];
    const float* wgt  = (const float*)d_in[1];
    const float* bias = (const float*)d_in[2];
    float* out        = (float*)d_out;

    // grid = ntile(12) * h(8) * mchunk(32) = 3072 blocks, 8 waves each
    dim3 grid(12 * 8 * 32);
    dim3 block(256);
    pattern_conv_wmma<<<grid, block, 0, stream>>>(x, wgt, bias, out);
}